// Attention_34754875359554
// MI455X (gfx1250) — compile-verified
//
#include <hip/hip_runtime.h>
#include <math.h>

typedef __attribute__((ext_vector_type(2))) float v2f;
typedef __attribute__((ext_vector_type(4))) float v4f;
typedef __attribute__((ext_vector_type(8))) float v8f;
typedef __attribute__((ext_vector_type(4))) int   v4i;

#define BATCH 64
#define ROWS  4096      // padded; valid prev rows = 4095 (row 4095 is `final`)
#define DIM   256
#define PITCH 68        // LDS row pitch (floats) for a 64-col slice: 68 % 64 = 4 -> conflict-free

#if defined(__AMDGCN__) && __has_builtin(__builtin_amdgcn_global_load_async_to_lds_b128)
#define USE_ASYNC_LDS 1
#else
#define USE_ASYNC_LDS 0
#endif

// Builtin signature (from hipcc diagnostic): param 0 is `v4i __device__*` == addrspace(1) v4i*.
#define TO_GLOBAL_V4I(p) ((__attribute__((address_space(1))) v4i*)(p))
#define TO_LDS_V4I(p)    ((__attribute__((address_space(3))) v4i*)(p))

// ---------------------------------------------------------------------------
// Kernel 1: scores[b][t] = dot(prev[b][t], final[b])  via V_WMMA_F32_16X16X4_F32
// Grid (32, 64), block 256 (8 waves). Each wave owns one 16-row tile:
//   tile = blockIdx.x*8 + wave, rows [tile*16, tile*16+16).
// A-frag (16x4 f32): lane L<16 -> (M=L,  K=kb+0..1); lane>=16 -> (M=L-16, K=kb+2..3)
// B-frag (4x16 f32): final[kb+k] replicated across all N -> same per-lane formula as A's K.
// D[m][n] is identical for all n; column N=0 lives in lanes 0 (M=0..7) and 16 (M=8..15).
// ---------------------------------------------------------------------------
__global__ __launch_bounds__(256) void scores_wmma_kernel(
    const float* __restrict__ x, float* __restrict__ scores)
{
  const int b    = blockIdx.y;
  const int wave = threadIdx.x >> 5;
  const int lane = threadIdx.x & 31;
  const int tile = blockIdx.x * 8 + wave;          // 0..255
  const int rowbase = tile * 16;
  const float* xb = x + (size_t)b * ROWS * DIM;

  __shared__ __align__(16) float sfinal[DIM];
  __shared__ __align__(16) float stage[8][16 * PITCH];

  if (threadIdx.x < 64) {
    *(v4f*)&sfinal[threadIdx.x * 4] = *(const v4f*)(xb + (size_t)4095 * DIM + threadIdx.x * 4);
  }
  __syncthreads();

  float* buf   = stage[wave];
  const int m  = lane & 15;
  const int kh = (lane >> 4) << 1;                 // 0 or 2

  v8f c = {};

  for (int slice = 0; slice < 4; ++slice) {
    const int colbase = slice * 64;
    // Stage 16 rows x 64 floats (16 KB tile sliced into 4 KB): coalesced async b128 per lane.
    #pragma unroll
    for (int i = 0; i < 8; ++i) {
      const int f4 = i * 32 + lane;                // 0..255 float4's
      const int r  = f4 >> 4;                      // 16 float4 per row-slice
      const int c4 = f4 & 15;
      const float* gp = xb + (size_t)(rowbase + r) * DIM + colbase + c4 * 4;
      float* lp = buf + r * PITCH + c4 * 4;
#if USE_ASYNC_LDS
      __builtin_amdgcn_global_load_async_to_lds_b128(
          TO_GLOBAL_V4I(const_cast<float*>(gp)), TO_LDS_V4I(lp), 0, 0);
#else
      *(v4f*)lp = *(const v4f*)gp;
#endif
    }
#if USE_ASYNC_LDS
#if __has_builtin(__builtin_amdgcn_s_wait_asynccnt)
    __builtin_amdgcn_s_wait_asynccnt(0);
#else
    asm volatile("s_wait_asynccnt 0" ::: "memory");
#endif
#endif
    // 16 K-chunks of 4 -> 64 total WMMAs per tile, accumulating into c.
    #pragma unroll
    for (int kc = 0; kc < 16; ++kc) {
      const int kb = kc * 4;
      v2f a  = *(const v2f*)(buf + m * PITCH + kb + kh);      // bank-conflict-free b64
      v2f bf = *(const v2f*)(&sfinal[colbase + kb + kh]);     // half-wave broadcast
      c = __builtin_amdgcn_wmma_f32_16x16x4_f32(
              /*neg_a=*/false, a, /*neg_b=*/false, bf,
              /*c_mod=*/(short)0, c, /*reuse_a=*/false, /*reuse_b=*/false);
    }
  }

  float* sc = scores + (size_t)b * ROWS + rowbase;
  if (lane == 0) {
    #pragma unroll
    for (int j = 0; j < 8; ++j) sc[j] = c[j];        // M = 0..7,  N = 0
  } else if (lane == 16) {
    #pragma unroll
    for (int j = 0; j < 8; ++j) sc[8 + j] = c[j];    // M = 8..15, N = 0
  }
}

// ---------------------------------------------------------------------------
// Kernel 2: per-batch softmax over t in [0,4095); normalizes scores in place
// into weights; scores[4095] forced to 0 so later kernels can run 4096 rows.
// ---------------------------------------------------------------------------
__global__ __launch_bounds__(256) void softmax_kernel(float* __restrict__ scores)
{
  const int b    = blockIdx.x;
  float* sc      = scores + (size_t)b * ROWS;
  const int tid  = threadIdx.x;
  const int lane = tid & 31, wave = tid >> 5;
  __shared__ float red[8];

  float v[16];
  float mx = -INFINITY;
  #pragma unroll
  for (int i = 0; i < 16; ++i) {
    const int t = i * 256 + tid;
    v[i] = (t < ROWS - 1) ? sc[t] : -INFINITY;
    mx = fmaxf(mx, v[i]);
  }
  #pragma unroll
  for (int off = 16; off >= 1; off >>= 1) mx = fmaxf(mx, __shfl_xor(mx, off, 32));
  if (lane == 0) red[wave] = mx;
  __syncthreads();
  float bm = red[0];
  #pragma unroll
  for (int i = 1; i < 8; ++i) bm = fmaxf(bm, red[i]);
  __syncthreads();

  float e[16];
  float s = 0.f;
  #pragma unroll
  for (int i = 0; i < 16; ++i) {
    const int t = i * 256 + tid;
    e[i] = (t < ROWS - 1) ? expf(v[i] - bm) : 0.f;
    s += e[i];
  }
  #pragma unroll
  for (int off = 16; off >= 1; off >>= 1) s += __shfl_xor(s, off, 32);
  if (lane == 0) red[wave] = s;
  __syncthreads();
  float tot = 0.f;
  #pragma unroll
  for (int i = 0; i < 8; ++i) tot += red[i];
  const float inv = 1.f / tot;
  #pragma unroll
  for (int i = 0; i < 16; ++i) {
    const int t = i * 256 + tid;
    sc[t] = e[i] * inv;                    // t == 4095 stores 0
  }
}

// ---------------------------------------------------------------------------
// Kernel 3: partial[b][rg][rsub][d] = sum over rows r0+rsub, step 4, of w[r]*prev[r][:].
// Grid (8, 64), block 256. Thread = (rsub = tid>>6, float4-slice d4 = tid&63):
// 64 threads span a 1 KB row (b128 loads), 4 rows in flight. w[r] is wave-uniform
// (lanes share r) -> scalar load. Fully coalesced, 4x fewer vmem ops than b32.
// ---------------------------------------------------------------------------
__global__ __launch_bounds__(256) void attn_partial_kernel(
    const float* __restrict__ x, const float* __restrict__ scores,
    float* __restrict__ partial)           // [BATCH][8][4][DIM] floats
{
  const int b = blockIdx.y, rg = blockIdx.x;
  const float* xb = x + (size_t)b * ROWS * DIM;
  const float* w  = scores + (size_t)b * ROWS;
  const int d4   = threadIdx.x & 63;
  const int rsub = threadIdx.x >> 6;       // 0..3
  const int rend = (rg + 1) * (ROWS / 8);
  v4f acc = {0.f, 0.f, 0.f, 0.f};
  #pragma unroll 4
  for (int r = rg * (ROWS / 8) + rsub; r < rend; r += 4) {
    const float wr = w[r];
    const v4f row = *(const v4f*)(xb + (size_t)r * DIM + d4 * 4);
    acc += row * wr;
  }
  *(v4f*)(partial + (((size_t)b * 8 + rg) * 4 + rsub) * DIM + d4 * 4) = acc;
}

// ---------------------------------------------------------------------------
// Kernel 4: out[b] = [ final | sum over 32 partials ]  (fixed order: deterministic)
// ---------------------------------------------------------------------------
__global__ __launch_bounds__(256) void finalize_kernel(
    const float* __restrict__ x, const float* __restrict__ partial,
    float* __restrict__ out)
{
  const int b = blockIdx.x, d = threadIdx.x;
  const float* xb = x + (size_t)b * ROWS * DIM;
  out[(size_t)b * 512 + d] = xb[(size_t)4095 * DIM + d];
  float s = 0.f;
  #pragma unroll
  for (int p = 0; p < 32; ++p) s += partial[((size_t)b * 32 + p) * DIM + d];
  out[(size_t)b * 512 + 256 + d] = s;
}

extern "C" void kernel_launch(void* const* d_in, const int* in_sizes, int n_in,
                              void* d_out, int out_size, void* d_ws, size_t ws_size,
                              hipStream_t stream)
{
  const float* x = (const float*)d_in[0];
  float* out     = (float*)d_out;
  float* scores  = (float*)d_ws;                         // 64*4096 f32    (1 MB)
  float* partial = scores + (size_t)BATCH * ROWS;        // 64*32*256 f32  (2 MB)

  scores_wmma_kernel<<<dim3(32, BATCH), 256, 0, stream>>>(x, scores);
  softmax_kernel    <<<BATCH,           256, 0, stream>>>(scores);
  attn_partial_kernel<<<dim3(8, BATCH), 256, 0, stream>>>(x, scores, partial);
  finalize_kernel   <<<BATCH,           256, 0, stream>>>(x, partial, out);
}